// GCN_58583353918035
// MI455X (gfx1250) — compile-verified
//
#include <hip/hip_runtime.h>
#include <math.h>

typedef __attribute__((ext_vector_type(2))) float v2f;
typedef __attribute__((ext_vector_type(8))) float v8f;

#define NN   10000      // nodes
#define NG   64         // graphs
#define DIM  256        // in/hidden dim
#define NC   10         // classes
#define NE   320000     // edges

// ---------------------------------------------------------------- utilities
__global__ void zero_kernel(float* __restrict__ p, int n) {
  int i = blockIdx.x * blockDim.x + threadIdx.x;
  if (i < n) p[i] = 0.0f;
}

__global__ void deg_init_kernel(float* __restrict__ deg) {
  int i = blockIdx.x * blockDim.x + threadIdx.x;
  if (i < NN) deg[i] = 1.0f;                      // self-loop weight
}

__global__ void deg_accum_kernel(const int* __restrict__ dst,
                                 float* __restrict__ deg) {
  int e = blockIdx.x * blockDim.x + threadIdx.x;
  if (e < NE) atomicAdd(&deg[dst[e]], 1.0f);
}

__global__ void dinv_kernel(float* __restrict__ deg) {
  int i = blockIdx.x * blockDim.x + threadIdx.x;
  if (i < NN) deg[i] = rsqrtf(deg[i]);            // deg >= 1 always
}

// ---------------------------------------------------------------- WMMA GEMM
// H[N,256] = relu?(X[N,256]) @ W[256,256]  (row-major, fp32, full precision)
// Block = 8 waves sharing one 64-col W panel staged in LDS (64 KB).
// Each wave computes a 16x64 strip: 4 f32 WMMA accumulators per A fragment.
// Grid: 4 col-groups x ceil(625/8) m-tile groups = 316 blocks.
__global__ __launch_bounds__(256)
void gemm_wmma_kernel(const float* __restrict__ X, const float* __restrict__ W,
                      float* __restrict__ H, int relu_in) {
  __shared__ float sW[DIM * 64];                  // W[:, ng*64 .. ng*64+63]

  const int wave = threadIdx.x >> 5;
  const int lane = threadIdx.x & 31;
  const int lo   = lane & 15;     // M (for A) / N (for B,C,D)
  const int hi   = lane >> 4;     // K-half select for A/B; M+8 select for C/D

  const int ng  = blockIdx.x & 3;                 // column group (64 cols)
  const int mt  = (blockIdx.x >> 2) * 8 + wave;   // m-tile for this wave

  // cooperative stage of the W panel: 256 rows x 64 cols, float4 per thread
  for (int i = threadIdx.x; i < DIM * 16; i += 256) {   // DIM*64/4 float4s
    int r  = i >> 4;
    int c4 = (i & 15) << 2;
    *(float4*)(sW + r * 64 + c4) = *(const float4*)(W + r * DIM + ng * 64 + c4);
  }
  __syncthreads();          // all waves reach barrier before any exits

  if (mt >= 625) return;    // wave-uniform exit (EXEC all-1s for WMMA)

  const float* xrow = X + (mt * 16 + lo) * DIM;
  v8f c0 = {}, c1 = {}, c2 = {}, c3 = {};

  for (int k = 0; k < DIM; k += 4) {
    const int ka = k + 2 * hi;
    // A fragment: lane lo holds X[row][ka], X[row][ka+1] (8B-aligned pair)
    v2f a = *(const v2f*)(xrow + ka);
    if (relu_in) {
      a.x = a.x > 0.0f ? a.x : 0.0f;
      a.y = a.y > 0.0f ? a.y : 0.0f;
    }
    // B fragments from LDS: b.x = W[ka][col], b.y = W[ka+1][col]
    const float* w0 = sW + ka * 64 + lo;
    const float* w1 = w0 + 64;
    v2f b0 = { w0[0],  w1[0]  };
    v2f b1 = { w0[16], w1[16] };
    v2f b2 = { w0[32], w1[32] };
    v2f b3 = { w0[48], w1[48] };
    c0 = __builtin_amdgcn_wmma_f32_16x16x4_f32(false, a, false, b0, (short)0, c0, false, false);
    c1 = __builtin_amdgcn_wmma_f32_16x16x4_f32(false, a, false, b1, (short)0, c1, false, false);
    c2 = __builtin_amdgcn_wmma_f32_16x16x4_f32(false, a, false, b2, (short)0, c2, false, false);
    c3 = __builtin_amdgcn_wmma_f32_16x16x4_f32(false, a, false, b3, (short)0, c3, false, false);
  }

  // D layout: VGPR r, lanes 0-15 -> M = r, lanes 16-31 -> M = 8 + r
  float* out = H + (mt * 16 + hi * 8) * DIM + ng * 64 + lo;
  #pragma unroll
  for (int r = 0; r < 8; ++r) {
    out[r * DIM + 0]  = c0[r];
    out[r * DIM + 16] = c1[r];
    out[r * DIM + 32] = c2[r];
    out[r * DIM + 48] = c3[r];
  }
}

// ------------------------------------------------- aggregation (self-loop + bias)
__global__ __launch_bounds__(256)
void init_out_kernel(const float* __restrict__ H, const float* __restrict__ dinv,
                     const float* __restrict__ bias, float* __restrict__ OUT) {
  int i = blockIdx.x * blockDim.x + threadIdx.x;   // over NN*DIM
  if (i >= NN * DIM) return;
  int node = i >> 8;
  int f    = i & 255;
  float dv = dinv[node];
  OUT[i] = dv * dv * H[i] + bias[f];
}

// edge scatter: one thread per (edge, 4-feature chunk); float4 gather + 4 f32 atomics
__global__ __launch_bounds__(256)
void scatter_kernel(const int* __restrict__ src, const int* __restrict__ dst,
                    const float* __restrict__ dinv,
                    const float* __restrict__ H, float* __restrict__ OUT) {
  long long t = (long long)blockIdx.x * blockDim.x + threadIdx.x;
  int e  = (int)(t >> 6);
  int ch = ((int)t & 63) << 2;
  if (e >= NE) return;
  int s = src[e], d = dst[e];
  float nrm = dinv[s] * dinv[d];
  float4 v = *(const float4*)(H + s * DIM + ch);
  float* o = OUT + d * DIM + ch;
  atomicAdd(o + 0, v.x * nrm);
  atomicAdd(o + 1, v.y * nrm);
  atomicAdd(o + 2, v.z * nrm);
  atomicAdd(o + 3, v.w * nrm);
}

// ---------------------------------------------------------------- pooling + head
__global__ void count_kernel(const int* __restrict__ batch, float* __restrict__ cnt) {
  int i = blockIdx.x * blockDim.x + threadIdx.x;
  if (i < NN) atomicAdd(&cnt[batch[i]], 1.0f);
}

__global__ __launch_bounds__(256)
void pool_kernel(const float* __restrict__ Hf, const int* __restrict__ batch,
                 float* __restrict__ pooled) {
  int t = blockIdx.x * blockDim.x + threadIdx.x;   // NN * 64 chunks
  int node = t >> 6;
  int ch   = (t & 63) << 2;
  if (node >= NN) return;
  int g = batch[node];
  float4 v = *(const float4*)(Hf + node * DIM + ch);
  float* o = pooled + g * DIM + ch;
  atomicAdd(o + 0, v.x > 0.0f ? v.x : 0.0f);       // ReLU fused
  atomicAdd(o + 1, v.y > 0.0f ? v.y : 0.0f);
  atomicAdd(o + 2, v.z > 0.0f ? v.z : 0.0f);
  atomicAdd(o + 3, v.w > 0.0f ? v.w : 0.0f);
}

__global__ void head_kernel(const float* __restrict__ pooled, const float* __restrict__ cnt,
                            const float* __restrict__ Wlin, const float* __restrict__ blin,
                            float* __restrict__ out) {
  int g = blockIdx.x;          // 64 graphs
  int c = threadIdx.x;         // 32 threads, c < 10 active for compute
  __shared__ float sl[NC];
  __shared__ float smx, slse;
  float inv = 1.0f / fmaxf(cnt[g], 1.0f);
  if (c < NC) {
    float acc = blin[c];
    for (int k = 0; k < DIM; ++k)
      acc += pooled[g * DIM + k] * inv * Wlin[k * NC + c];
    sl[c] = acc;
  }
  __syncthreads();
  if (c == 0) {
    float m = sl[0];
    for (int i = 1; i < NC; ++i) m = fmaxf(m, sl[i]);
    float s = 0.0f;
    for (int i = 0; i < NC; ++i) s += expf(sl[i] - m);
    smx = m; slse = logf(s);
  }
  __syncthreads();
  if (c < NC) out[g * NC + c] = sl[c] - smx - slse;
}

// ---------------------------------------------------------------- driver
extern "C" void kernel_launch(void* const* d_in, const int* in_sizes, int n_in,
                              void* d_out, int out_size, void* d_ws, size_t ws_size,
                              hipStream_t stream) {
  const float* x    = (const float*)d_in[0];
  const float* W1   = (const float*)d_in[1];
  const float* b1   = (const float*)d_in[2];
  const float* W2   = (const float*)d_in[3];
  const float* b2   = (const float*)d_in[4];
  const float* W3   = (const float*)d_in[5];
  const float* b3   = (const float*)d_in[6];
  const float* Wlin = (const float*)d_in[7];
  const float* blin = (const float*)d_in[8];
  const int*   eidx = (const int*)d_in[9];      // [2, NE] flat
  const int*   batch= (const int*)d_in[10];
  float* out = (float*)d_out;

  const int* src = eidx;
  const int* dst = eidx + NE;

  // workspace layout (floats)
  float* ws     = (float*)d_ws;
  float* dinv   = ws;                   // NN (deg -> dinv in place)
  float* bufA   = ws + 16384;           // NN*DIM  (GEMM output H)
  float* bufB   = bufA + NN * DIM;      // NN*DIM  (aggregated output)
  float* pooled = bufB + NN * DIM;      // NG*DIM
  float* cnt    = pooled + NG * DIM;    // NG

  const int B = 256;
  // degrees -> dinv
  deg_init_kernel <<<(NN + B - 1) / B, B, 0, stream>>>(dinv);
  deg_accum_kernel<<<(NE + B - 1) / B, B, 0, stream>>>(dst, dinv);
  dinv_kernel     <<<(NN + B - 1) / B, B, 0, stream>>>(dinv);
  // zero pooled sums + counts
  zero_kernel     <<<(NG * DIM + NG + B - 1) / B, B, 0, stream>>>(pooled, NG * DIM + NG);

  const int gemmBlocks    = 4 * ((625 + 7) / 8);         // 316
  const int initBlocks    = (NN * DIM) / B;              // 10000
  const int scatterBlocks = (NE * 64) / B;               // 80000

  // layer 1: input x (no relu)
  gemm_wmma_kernel<<<gemmBlocks, B, 0, stream>>>(x, W1, bufA, 0);
  init_out_kernel <<<initBlocks, B, 0, stream>>>(bufA, dinv, b1, bufB);
  scatter_kernel  <<<scatterBlocks, B, 0, stream>>>(src, dst, dinv, bufA, bufB);
  // layer 2: input relu(bufB)
  gemm_wmma_kernel<<<gemmBlocks, B, 0, stream>>>(bufB, W2, bufA, 1);
  init_out_kernel <<<initBlocks, B, 0, stream>>>(bufA, dinv, b2, bufB);
  scatter_kernel  <<<scatterBlocks, B, 0, stream>>>(src, dst, dinv, bufA, bufB);
  // layer 3
  gemm_wmma_kernel<<<gemmBlocks, B, 0, stream>>>(bufB, W3, bufA, 1);
  init_out_kernel <<<initBlocks, B, 0, stream>>>(bufA, dinv, b3, bufB);
  scatter_kernel  <<<scatterBlocks, B, 0, stream>>>(src, dst, dinv, bufA, bufB);

  // mean pool (relu fused) + head
  count_kernel<<<(NN + B - 1) / B, B, 0, stream>>>(batch, cnt);
  pool_kernel <<<(NN * 64) / B, B, 0, stream>>>(bufB, batch, pooled);
  head_kernel <<<NG, 32, 0, stream>>>(pooled, cnt, Wlin, blin, out);
}